// graphNetwork_nodesOnly_32392643346733
// MI455X (gfx1250) — compile-verified
//
#include <hip/hip_runtime.h>
#include <hip/hip_bf16.h>
#include <math.h>

typedef __attribute__((ext_vector_type(2))) float v2f;
typedef __attribute__((ext_vector_type(4))) float v4f;
typedef __attribute__((ext_vector_type(8))) float v8f;

#define CH 64          // hidden channels
#define CIN 128        // input channels
#define COUT 1024      // output channels
#define H2 0.01f       // h*h with h = 0.1

// ---------------------------------------------------------------------------
// gcn_norm helpers: deg (with self loop), dinv = rsqrt(deg), w2 = (di*dj)^2
// ---------------------------------------------------------------------------
__global__ __launch_bounds__(256) void k_deg_init(float* deg, int N) {
    int n = blockIdx.x * blockDim.x + threadIdx.x;
    if (n < N) deg[n] = 1.0f;   // self loop contributes 1 to deg
}

__global__ __launch_bounds__(256) void k_deg_acc(float* deg, const int* __restrict__ J, int E) {
    int e = blockIdx.x * blockDim.x + threadIdx.x;
    if (e < E) atomicAdd(&deg[J[e]], 1.0f);
}

__global__ __launch_bounds__(256) void k_dinv(float* deg, int N) {
    int n = blockIdx.x * blockDim.x + threadIdx.x;
    if (n < N) deg[n] = rsqrtf(deg[n]);   // deg >= 1 always (self loops)
}

__global__ __launch_bounds__(256) void k_w2(const float* __restrict__ dinv,
                                            const int* __restrict__ I,
                                            const int* __restrict__ J,
                                            float* __restrict__ w2, int E) {
    int e = blockIdx.x * blockDim.x + threadIdx.x;
    if (e < E) {
        float w = dinv[I[e]] * dinv[J[e]];
        w2[e] = w * w;   // w applied in node_grad AND edge_div
    }
}

// ---------------------------------------------------------------------------
// Incidence CSR build: cnt -> exclusive scan -> fill (neighbor, weight) pairs.
// Contribution of an edge (i,j,w2) to node i is w2*(x_i-x_j), to node j is
// w2*(x_j-x_i): symmetric, no sign needed in the incidence list.
// ---------------------------------------------------------------------------
__global__ __launch_bounds__(256) void k_cnt(int* cnt, const int* __restrict__ I,
                                             const int* __restrict__ J, int E) {
    int e = blockIdx.x * blockDim.x + threadIdx.x;
    if (e < E) {
        atomicAdd(&cnt[I[e]], 1);
        atomicAdd(&cnt[J[e]], 1);
    }
}

// Single-block exclusive scan over N elements (N ~ 50000): LDS Hillis-Steele
// per 1024-chunk with a running carry. rowstart has N+1 entries.
__global__ __launch_bounds__(1024) void k_scan(const int* __restrict__ cnt,
                                               int* __restrict__ rowstart, int N) {
    __shared__ int sh[1024];
    __shared__ int carry;
    if (threadIdx.x == 0) carry = 0;
    __syncthreads();
    for (int base = 0; base < N; base += 1024) {
        int i = base + threadIdx.x;
        int v = (i < N) ? cnt[i] : 0;
        sh[threadIdx.x] = v;
        __syncthreads();
        for (int ofs = 1; ofs < 1024; ofs <<= 1) {
            int t = (threadIdx.x >= (unsigned)ofs) ? sh[threadIdx.x - ofs] : 0;
            __syncthreads();
            sh[threadIdx.x] += t;
            __syncthreads();
        }
        int incl = sh[threadIdx.x];
        if (i < N) rowstart[i] = carry + incl - v;   // exclusive
        __syncthreads();                              // all reads of carry done
        if (threadIdx.x == 1023) carry += sh[1023];
        __syncthreads();
    }
    if (threadIdx.x == 0) rowstart[N] = carry;        // total = 2E
}

__global__ __launch_bounds__(256) void k_fill(const int* __restrict__ I,
                                              const int* __restrict__ J,
                                              const float* __restrict__ w2,
                                              int* cur, int* __restrict__ adjn,
                                              float* __restrict__ adjw, int E) {
    int e = blockIdx.x * blockDim.x + threadIdx.x;
    if (e < E) {
        int i = I[e], j = J[e];
        float w = w2[e];
        int p = atomicAdd(&cur[i], 1);
        adjn[p] = j; adjw[p] = w;
        int q = atomicAdd(&cur[j], 1);
        adjn[q] = i; adjw[q] = w;
    }
}

// ---------------------------------------------------------------------------
// Laplacian gather (no atomics): one 64-thread block per node, thread = channel.
// lap[n][c] = (sum_k w_k) * x[n][c] - sum_k w_k * x[m_k][c]
// ---------------------------------------------------------------------------
__global__ __launch_bounds__(64) void k_lap(const float* __restrict__ x,
                                            float* __restrict__ lap,
                                            const int* __restrict__ rowstart,
                                            const int* __restrict__ adjn,
                                            const float* __restrict__ adjw) {
    const int n = blockIdx.x;
    const int c = threadIdx.x;
    const int s = rowstart[n];
    const int t = rowstart[n + 1];
    const float xc = x[(size_t)n * CH + c];
    float acc = 0.0f, wsum = 0.0f;
    for (int k = s; k < t; ++k) {
        const int m = adjn[k];       // block-uniform
        const float wv = adjw[k];    // block-uniform
        wsum += wv;
        acc += wv * x[(size_t)m * CH + c];   // 256B contiguous per block
    }
    lap[(size_t)n * CH + c] = wsum * xc - acc;
}

// ---------------------------------------------------------------------------
// Opening GEMM: x[node][o] = relu(sum_c K1[o][c] * xn[c][node]).
// WMMA tile orientation: M = output channels (A from K1), N = nodes (B from xn).
// D lane element (vgpr r): channel o0 + r + 8*hi, node n0 + lo -> each lane owns
// 8 contiguous floats of a node row => two b128 stores.
// ---------------------------------------------------------------------------
__global__ __launch_bounds__(128) void k_open(const float* __restrict__ xn,   // [128][N]
                                              const float* __restrict__ K1,  // [64][128]
                                              float* __restrict__ x,         // [N][64]
                                              float* __restrict__ xold,      // [N][64]
                                              int N) {
    const int lane = threadIdx.x & 31;
    const int wave = threadIdx.x >> 5;
    const int lo = lane & 15, hi = lane >> 4;
    const int n0 = blockIdx.x * 16;
    const int o0 = wave * 16;

    v8f acc = {};
    for (int k0 = 0; k0 < CIN; k0 += 4) {
        const int ka = k0 + 2 * hi;
        // A[m][k] = K1[o0+m][k]: contiguous aligned float2
        v2f a = *(const v2f*)(K1 + (size_t)(o0 + lo) * CIN + ka);
        // B[k][n] = xn[k][n0+n]: strided scalar loads (xn read once -> NT)
        v2f b;
        b.x = __builtin_nontemporal_load(xn + (size_t)ka * N + n0 + lo);
        b.y = __builtin_nontemporal_load(xn + (size_t)(ka + 1) * N + n0 + lo);
        acc = __builtin_amdgcn_wmma_f32_16x16x4_f32(false, a, false, b,
                                                    (short)0, acc, false, false);
    }
    v8f v;
#pragma unroll
    for (int r = 0; r < 8; ++r) {
        float f = acc[r];
        v[r] = f > 0.0f ? f : 0.0f;           // relu
    }
    const size_t base = (size_t)(n0 + lo) * CH + o0 + 8 * hi;  // 32B aligned
    *(v4f*)(x + base)        = *(v4f*)&v;
    *(v4f*)(x + base + 4)    = *((v4f*)&v + 1);
    *(v4f*)(xold + base)     = *(v4f*)&v;
    *(v4f*)(xold + base + 4) = *((v4f*)&v + 1);
}

// ---------------------------------------------------------------------------
// Step GEMM + leapfrog: xnew = 2x - xold - h^2 * tanh(Ks * lap_row)
// M = output channels (A from Ks), N = nodes (B from lap, contiguous float2).
// ---------------------------------------------------------------------------
__global__ __launch_bounds__(128) void k_step(const float* __restrict__ lap,  // [N][64]
                                              const float* __restrict__ Ks,   // [64][64]
                                              const float* __restrict__ x,    // [N][64]
                                              const float* __restrict__ xold, // [N][64]
                                              float* __restrict__ xnew,       // [N][64]
                                              int N) {
    const int lane = threadIdx.x & 31;
    const int wave = threadIdx.x >> 5;
    const int lo = lane & 15, hi = lane >> 4;
    const int n0 = blockIdx.x * 16;
    const int o0 = wave * 16;

    v8f acc = {};
#pragma unroll
    for (int k0 = 0; k0 < CH; k0 += 4) {
        const int ka = k0 + 2 * hi;
        v2f a = *(const v2f*)(Ks + (size_t)(o0 + lo) * CH + ka);
        v2f b = *(const v2f*)(lap + (size_t)(n0 + lo) * CH + ka);
        acc = __builtin_amdgcn_wmma_f32_16x16x4_f32(false, a, false, b,
                                                    (short)0, acc, false, false);
    }
    const size_t base = (size_t)(n0 + lo) * CH + o0 + 8 * hi;  // 32B aligned
    v4f xc0 = *(const v4f*)(x + base),     xc1 = *(const v4f*)(x + base + 4);
    v4f xo0 = *(const v4f*)(xold + base),  xo1 = *(const v4f*)(xold + base + 4);
    v4f r0, r1;
#pragma unroll
    for (int r = 0; r < 4; ++r) {
        r0[r] = 2.0f * xc0[r] - xo0[r] - H2 * tanhf(acc[r]);
        r1[r] = 2.0f * xc1[r] - xo1[r] - H2 * tanhf(acc[r + 4]);
    }
    *(v4f*)(xnew + base)     = r0;
    *(v4f*)(xnew + base + 4) = r1;
}

// ---------------------------------------------------------------------------
// Closing GEMM: out[o][n] = sum_c Kc[o][c] * x[n][c].
// M = output channels (A from Kc), N = nodes. Per-vgpr store: lanes 0-15 hit
// 16 consecutive floats of an out row -> coalesced; NT (write-once, > L2).
// ---------------------------------------------------------------------------
__global__ __launch_bounds__(128) void k_close(const float* __restrict__ x,   // [N][64]
                                               const float* __restrict__ Kc,  // [1024][64]
                                               float* __restrict__ out,       // [1024][N]
                                               int N) {
    const int lane = threadIdx.x & 31;
    const int wave = threadIdx.x >> 5;
    const int lo = lane & 15, hi = lane >> 4;
    const int n0 = blockIdx.x * 16;
    const int o0 = (blockIdx.y * 4 + wave) * 16;

    v8f acc = {};
#pragma unroll
    for (int k0 = 0; k0 < CH; k0 += 4) {
        const int ka = k0 + 2 * hi;
        v2f a = *(const v2f*)(Kc + (size_t)(o0 + lo) * CH + ka);
        v2f b = *(const v2f*)(x + (size_t)(n0 + lo) * CH + ka);
        acc = __builtin_amdgcn_wmma_f32_16x16x4_f32(false, a, false, b,
                                                    (short)0, acc, false, false);
    }
#pragma unroll
    for (int r = 0; r < 8; ++r) {
        const int o = o0 + r + 8 * hi;
        __builtin_nontemporal_store(acc[r], out + (size_t)o * N + n0 + lo);
    }
}

// ---------------------------------------------------------------------------
extern "C" void kernel_launch(void* const* d_in, const int* in_sizes, int n_in,
                              void* d_out, int out_size, void* d_ws, size_t ws_size,
                              hipStream_t stream) {
    const float* xn  = (const float*)d_in[0];  // [1,128,N]
    const int*   ei  = (const int*)d_in[1];    // [E]
    const int*   ej  = (const int*)d_in[2];    // [E]
    const float* K1  = (const float*)d_in[3];  // [64,128]
    const float* KN1 = (const float*)d_in[4];  // [4,64,64]
    const float* Kc  = (const float*)d_in[5];  // [1024,64]
    float* out = (float*)d_out;                // [1024,N]

    const int N = in_sizes[0] / CIN;           // 50000 (divisible by 16)
    const int E = in_sizes[1];                 // 800000

    // Workspace carve-out (256B aligned regions)
    char* ws = (char*)d_ws;
    size_t off = 0;
    auto carve = [&](size_t bytes) -> void* {
        void* p = ws + off;
        off = (off + bytes + 255) & ~(size_t)255;
        return p;
    };
    float* deg      = (float*)carve((size_t)N * 4);        // deg -> dinv in place
    float* w2       = (float*)carve((size_t)E * 4);
    int*   cnt      = (int*)carve((size_t)N * 4);
    int*   rowstart = (int*)carve((size_t)(N + 1) * 4);
    int*   cur      = (int*)carve((size_t)N * 4);
    int*   adjn     = (int*)carve((size_t)2 * E * 4);
    float* adjw     = (float*)carve((size_t)2 * E * 4);
    float* XA       = (float*)carve((size_t)N * CH * 4);
    float* XB       = (float*)carve((size_t)N * CH * 4);
    float* XC       = (float*)carve((size_t)N * CH * 4);
    float* lap      = (float*)carve((size_t)N * CH * 4);

    const int eb = (E + 255) / 256;
    const int nb = (N + 255) / 256;

    // --- gcn_norm ---
    k_deg_init<<<nb, 256, 0, stream>>>(deg, N);
    k_deg_acc<<<eb, 256, 0, stream>>>(deg, ej, E);
    k_dinv<<<nb, 256, 0, stream>>>(deg, N);
    k_w2<<<eb, 256, 0, stream>>>(deg, ei, ej, w2, E);

    // --- incidence CSR build (once per call; removes all per-step atomics) ---
    hipMemsetAsync(cnt, 0, (size_t)N * 4, stream);
    k_cnt<<<eb, 256, 0, stream>>>(cnt, ei, ej, E);
    k_scan<<<1, 1024, 0, stream>>>(cnt, rowstart, N);
    hipMemcpyAsync(cur, rowstart, (size_t)N * 4, hipMemcpyDeviceToDevice, stream);
    k_fill<<<eb, 256, 0, stream>>>(ei, ej, w2, cur, adjn, adjw, E);

    const int ntiles = N / 16;   // 3125 full 16-node tiles

    // --- opening 1x1 conv + relu ---
    k_open<<<ntiles, 128, 0, stream>>>(xn, K1, XA, XB, N);

    // --- 4 leapfrog steps with pointer rotation (xcur, xold, xnew) ---
    float* xcur = XA; float* xold = XB; float* xnew = XC;
    for (int s = 0; s < 4; ++s) {
        k_lap<<<N, 64, 0, stream>>>(xcur, lap, rowstart, adjn, adjw);
        k_step<<<ntiles, 128, 0, stream>>>(lap, KN1 + (size_t)s * CH * CH,
                                           xcur, xold, xnew, N);
        float* t = xold; xold = xcur; xcur = xnew; xnew = t;
    }

    // --- closing projection ---
    dim3 cgrid(ntiles, COUT / CH);   // (3125, 16)
    k_close<<<cgrid, 128, 0, stream>>>(xcur, Kc, out, N);
}